// Relation_37718402794086
// MI455X (gfx1250) — compile-verified
//
#include <hip/hip_runtime.h>

// Dilated neighborhood attention, shapes (4,256,64,64), 8 heads, hdim 32,
// 5x5 window, dilation 3, zero-pad 6. Head-averaged attention applied to all
// channels (einsum 'nckhw,nakhw->nchw' / NHEADS).
#define NB    4
#define CCH   256
#define HH    64
#define WW    64
#define NH    8
#define HD    32
#define KS    5
#define DIL   3
#define PADR  6
#define K2    25
#define TQ    16           // queries per block (one row segment)
#define STRIP 28           // union key strip width: TQ + 2*PADR
#define SPAD  33           // padded LDS row stride (bank-conflict free; TDM lds_inc)

typedef __attribute__((ext_vector_type(2))) float v2f;   // A/B frag, wmma f32 16x16x4
typedef __attribute__((ext_vector_type(8))) float v8f;   // C/D frag

#if defined(__AMDGCN__) && __has_builtin(__builtin_amdgcn_tensor_load_to_lds) && \
    __has_builtin(__builtin_amdgcn_s_wait_tensorcnt)
#define USE_TDM 1
#else
#define USE_TDM 0
#endif

#if USE_TDM
typedef unsigned int u32x4 __attribute__((ext_vector_type(4)));
typedef int          i32x4 __attribute__((ext_vector_type(4)));
typedef int          i32x8 __attribute__((ext_vector_type(8)));

// Async-DMA one strip: cnt contiguous floats per row, HD=32 rows at global
// stride H*W elements, into LDS rows of stride SPAD floats (D# iterate mode).
__device__ __forceinline__ void tdm_load_strip(const float* gsrc, unsigned lds_byte,
                                               int cnt) {
    unsigned long long ga = (unsigned long long)(const void*)gsrc;
    u32x4 g0;
    g0.x = 1u;                                    // D# count=1, user-mode load
    g0.y = lds_byte;                              // lds_addr (bytes, wave-relative)
    g0.z = (unsigned)(ga & 0xFFFFFFFFu);          // global_addr[31:0]
    g0.w = (unsigned)((ga >> 32) & 0x01FFFFFFu)   // global_addr[56:32]
         | 0x80000000u;                           // type = 2 ("image")
    i32x8 g1;
    g1[0] = 0x000A0000;                           // data_size=4B (2), iterate_enable
    g1[1] = (cnt & 0xFFFF) << 16;                 // tensor_dim0[15:0]
    g1[2] = 0x00010000;                           // tensor_dim0[31:16]=0, tensor_dim1=1
    g1[3] = (cnt & 0xFFFF) << 16;                 // tensor_dim1[31:16]=0, tile_dim0=cnt
    g1[4] = 1;                                    // tile_dim1=1, tile_dim2=0
    g1[5] = WW;                                   // tensor_dim0_stride (tile_dim1==1)
    g1[6] = 0;
    g1[7] = 0;
    i32x4 g2;
    g2[0] = 1;                                    // tensor_dim2 (unused)
    g2[1] = SPAD;                                 // lds_addr_increment / iteration
    g2[2] = HH * WW;                              // global_addr_increment / iteration
    g2[3] = (HD - 1) << 16;                       // iterate_count = 31 -> 32 rows
    i32x4 g3 = {0, 0, 0, 0};
    i32x8 g4 = {0, 0, 0, 0, 0, 0, 0, 0};          // 6-arg toolchain: zero-filled tail
    __builtin_amdgcn_tensor_load_to_lds(g0, g1, g2, g3, g4, 0);
}
#endif

// Stage one zero-padded strip into sreg[HD][SPAD] (layout [channel][strip col]).
// gbase points at (n, c0, row_clamped, lo_w). Wave-uniform sOff/cnt; rok==false
// means the whole row is outside -> all STRIP columns zero. Columns 28..31 are
// never consumed and stay don't-care.
__device__ __forceinline__ void stage_strip(float* sreg, const float* gbase,
                                            int ln, int sOff, int cnt, bool rok,
                                            size_t chStride) {
#if USE_TDM
    const int lim = rok ? sOff : STRIP;           // rok=false -> zero everything
    for (int s = 0; s < lim; ++s)                 sreg[ln * SPAD + s] = 0.0f;
    if (rok)
        for (int s = sOff + cnt; s < STRIP; ++s)  sreg[ln * SPAD + s] = 0.0f;
    if (rok)
        tdm_load_strip(gbase, (unsigned)(unsigned long long)(void*)sreg
                               + (unsigned)(sOff * 4), cnt);
    __builtin_amdgcn_s_wait_tensorcnt(0);
#else
    const int s = ln;                             // lane owns one strip column
    const bool ok = rok && (s >= sOff) && (s < sOff + cnt);
    const float* gp = gbase + (s - sOff);
    #pragma unroll 4
    for (int d = 0; d < HD; ++d)
        sreg[d * SPAD + s] = ok ? gp[(size_t)d * chStride] : 0.0f;
#endif
}

// LDS (floats): sKV [NH][32][SPAD] K-strip overlaid with V-strip   = 8448
//               sS  [NH][16][SPAD] scores -> att -> out transpose  = 4224
//               sWall [KS][16][32] head-averaged attention (A mat) = 2560
// total 15232 floats = 60928 B static.

__global__ void __launch_bounds__(256)
natt_fused(const float* __restrict__ Vg,
           const float* __restrict__ Kg,
           const float* __restrict__ Qg,
           float* __restrict__ outg)
{
    __shared__ float sKV[NH * 32 * SPAD];
    __shared__ float sS [NH * 16 * SPAD];
    __shared__ float sWall[KS * 16 * 32];

    const int tid  = threadIdx.x;
    const int a    = tid >> 5;      // wave id == head id
    const int ln   = tid & 31;
    const int lo   = ln & 15;       // M (query) or N (col) within a 16x16 tile
    const int half = ln >> 4;       // selects the K-pair held by this lane half

    int bid = blockIdx.x;
    const int w0 = (bid & 3) * TQ;  bid >>= 2;
    const int h  = bid & 63;        bid >>= 6;
    const int n  = bid;

    float* sKa = sKV + a * 32 * SPAD;
    float* sSa = sS  + a * 16 * SPAD;

    const size_t chStride = (size_t)HH * WW;

    // clipped strip extent (block-uniform): strip col s maps to w = w0-6+s
    const int lo_w = (w0 - PADR > 0) ? (w0 - PADR) : 0;
    const int hi_w = (w0 + TQ - 1 + PADR < WW - 1) ? (w0 + TQ - 1 + PADR) : (WW - 1);
    const int cnt  = hi_w - lo_w + 1;
    const int sOff = lo_w - (w0 - PADR);

    // ---------------- Q A-fragments in registers for all 5 rows ----------------
    // A(16x4): lane M = lo; .x/.y hold K = 2*half, 2*half+1.
    v2f aq[8];
    {
        const float* qb = Qg + (((size_t)n * CCH + a * HD) * HH + h) * WW + w0 + lo;
        #pragma unroll
        for (int ks = 0; ks < 8; ++ks) {
            const int d0 = ks * 4 + half * 2;
            aq[ks].x = qb[(size_t)d0 * chStride];
            aq[ks].y = qb[(size_t)(d0 + 1) * chStride];
        }
    }

    // ------------- stage 1: S_r(16x28) = Q(16x32) x K_strip_r(32x28) -------------
    float sc[K2];
    #pragma unroll
    for (int r = 0; r < KS; ++r) {
        const int row  = h + (r - KS / 2) * DIL;
        const bool rok = (row >= 0) && (row < HH);
        const int rcl  = rok ? row : 0;
        stage_strip(sKa, Kg + (((size_t)n * CCH + a * HD) * HH + rcl) * WW + lo_w,
                    ln, sOff, cnt, rok, chStride);
        // D[p][s] = sum_d Q[p][d] * Kstrip[d][s]; two 16-wide N tiles
        v8f acc0 = {}; v8f acc1 = {};
        #pragma unroll
        for (int ks = 0; ks < 8; ++ks) {
            const int d0 = ks * 4 + half * 2;
            v2f b0, b1;
            b0.x = sKa[d0 * SPAD + lo];
            b0.y = sKa[(d0 + 1) * SPAD + lo];
            b1.x = sKa[d0 * SPAD + 16 + lo];
            b1.y = sKa[(d0 + 1) * SPAD + 16 + lo];
            acc0 = __builtin_amdgcn_wmma_f32_16x16x4_f32(false, aq[ks], false, b0,
                                                         (short)0, acc0, false, false);
            acc1 = __builtin_amdgcn_wmma_f32_16x16x4_f32(false, aq[ks], false, b1,
                                                         (short)0, acc1, false, false);
        }
        // stage D, then query p extracts its 5 strip columns s = p + 3j
        #pragma unroll
        for (int v = 0; v < 8; ++v) {
            const int M = v + half * 8;
            sSa[M * SPAD + lo]      = acc0[v];
            sSa[M * SPAD + 16 + lo] = acc1[v];
        }
        #pragma unroll
        for (int j = 0; j < 5; ++j)
            sc[r * 5 + j] = sSa[lo * SPAD + lo + 3 * j];
    }

    // ---------------- softmax over 25 window positions (in-lane) ----------------
    float m = sc[0];
    #pragma unroll
    for (int i = 1; i < K2; ++i) m = fmaxf(m, sc[i]);
    float sum = 0.0f;
    #pragma unroll
    for (int i = 0; i < K2; ++i) { sc[i] = __expf(sc[i] - m); sum += sc[i]; }
    const float inv = 1.0f / sum;
    if (half == 0) {
        #pragma unroll
        for (int i = 0; i < K2; ++i) sSa[lo * SPAD + i] = sc[i] * inv;
    }
    __syncthreads();

    // ---- W̄_r(16x32): head-averaged attention scattered into the strip (A mat) ----
    for (int idx = tid; idx < KS * 16 * 32; idx += 256) {
        const int r = idx >> 9;
        const int p = (idx >> 5) & 15;
        const int s = idx & 31;
        float v = 0.0f;
        const int t = s - p;
        if (t >= 0 && t <= 12 && (t % 3) == 0) {
            const int k = r * 5 + (t / 3);
            float acc = 0.0f;
            #pragma unroll
            for (int aa = 0; aa < NH; ++aa)
                acc += sS[aa * 16 * SPAD + p * SPAD + k];
            v = acc * (1.0f / NH);
        }
        sWall[idx] = v;
    }
    __syncthreads();

    // ----- stage 2: out(16 x 32c) += W̄_r(16x28) x V_strip_r(28x32c), per wave -----
    float* sVa = sKa;                   // overlay: K region reused for V strips
    v8f o0 = {}; v8f o1 = {};
    #pragma unroll
    for (int r = 0; r < KS; ++r) {
        const int row  = h + (r - KS / 2) * DIL;
        const bool rok = (row >= 0) && (row < HH);
        const int rcl  = rok ? row : 0;
        stage_strip(sVa, Vg + (((size_t)n * CCH + a * HD) * HH + rcl) * WW + lo_w,
                    ln, sOff, cnt, rok, chStride);
        #pragma unroll
        for (int ks = 0; ks < 7; ++ks) {          // K-dim = 28 strip positions
            const int s0 = ks * 4 + half * 2;
            v2f aw, b0, b1;
            aw.x = sWall[(r << 9) + (lo << 5) + s0];
            aw.y = sWall[(r << 9) + (lo << 5) + s0 + 1];
            b0.x = sVa[lo * SPAD + s0];             // B[K=s][N=c]: V strip [c][s]
            b0.y = sVa[lo * SPAD + s0 + 1];
            b1.x = sVa[(16 + lo) * SPAD + s0];
            b1.y = sVa[(16 + lo) * SPAD + s0 + 1];
            o0 = __builtin_amdgcn_wmma_f32_16x16x4_f32(false, aw, false, b0,
                                                       (short)0, o0, false, false);
            o1 = __builtin_amdgcn_wmma_f32_16x16x4_f32(false, aw, false, b1,
                                                       (short)0, o1, false, false);
        }
    }

    // ------------- output: transpose via LDS, then float4 stores -------------
    #pragma unroll
    for (int v = 0; v < 8; ++v) {
        const int wl = v + half * 8;
        sSa[lo * 16 + wl]        = o0[v];       // c_local = lo
        sSa[(16 + lo) * 16 + wl] = o1[v];       // c_local = 16 + lo
    }
    {
        const int c = a * HD + ln;              // lane owns one channel row
        float* ob = outg + (((size_t)n * CCH + c) * HH + h) * WW + w0;
        const float* sp = sSa + ln * 16;
        float4 x0 = make_float4(sp[0],  sp[1],  sp[2],  sp[3]);
        float4 x1 = make_float4(sp[4],  sp[5],  sp[6],  sp[7]);
        float4 x2 = make_float4(sp[8],  sp[9],  sp[10], sp[11]);
        float4 x3 = make_float4(sp[12], sp[13], sp[14], sp[15]);
        *(float4*)(ob + 0)  = x0;
        *(float4*)(ob + 4)  = x1;
        *(float4*)(ob + 8)  = x2;
        *(float4*)(ob + 12) = x3;
    }
}

extern "C" void kernel_launch(void* const* d_in, const int* in_sizes, int n_in,
                              void* d_out, int out_size, void* d_ws, size_t ws_size,
                              hipStream_t stream) {
    const float* V = (const float*)d_in[0];
    const float* K = (const float*)d_in[1];
    const float* Q = (const float*)d_in[2];
    float* out = (float*)d_out;
    (void)in_sizes; (void)n_in; (void)out_size; (void)d_ws; (void)ws_size;
    dim3 grid(NB * HH * (WW / TQ));   // 4 * 64 * 4 = 1024 blocks, 8 waves each
    natt_fused<<<grid, dim3(256), 0, stream>>>(V, K, Q, out);
}